// CentroidLayer_34084860461076
// MI455X (gfx1250) — compile-verified
//
#include <hip/hip_runtime.h>
#include <hip/hip_bf16.h>
#include <math.h>

// ---------------------------------------------------------------------------
// CentroidLayer Karcher-flow update, fp32 WMMA formulation for gfx1250.
//   All matrix functions (sqrt/invsqrt/log/exp of 32x32 SPD) use inverse-free
//   Newton-Schulz / Taylor iterations -> batched 32x32 fp32 GEMM on the matrix
//   cores (V_WMMA_F32_16X16X4_F32; fp32 is the precision floor for these
//   iterations, so the f32 WMMA path is the right instruction).
//   One 128-thread block (4 wave32) per 32x32 problem; wave w owns output
//   tile (w>>1, w&1); LDS matrices padded to stride 33 (bank-conflict free).
//   mmf<AF_A,AF_B>() computes (sa*A + da*I) @ (sb*B + db*I); the affine
//   transforms are applied at fragment construction ONLY when the template
//   flag is set (compile-time), so plain GEMMs carry zero VALU between the
//   ds_loads and the WMMA chain (avoids the un-foldable "x + 0.0f" issue).
//   Fragments are preloaded before the WMMA chain: loads pipeline into a
//   single dscnt wait per GEMM instead of one wait per K-step.
// ---------------------------------------------------------------------------

typedef float v2f __attribute__((ext_vector_type(2)));
typedef float v8f __attribute__((ext_vector_type(8)));

#define LDA 33
#define MSZ (32 * LDA)
#define ETA_OVER_B (0.01f / 1024.0f)

// D = (sa*A + da*I) @ (sb*B + db*I), 32x32 fp32 in LDS (row stride LDA).
// In-place safe (D may alias A/B): pre-write barrier orders all reads first.
template <int AF_A, int AF_B>
__device__ __forceinline__ void mmf(const float* A, float sa, float da,
                                    const float* B, float sb, float db,
                                    float* D) {
  const int lane = threadIdx.x & 31;
  const int wave = threadIdx.x >> 5;
  const int ti = (wave >> 1) << 4;   // row tile base
  const int tj = (wave & 1) << 4;    // col tile base
  const int mr = lane & 15;
  const int kh = (lane >> 4) << 1;   // K sub-offset 0 or 2 (ISA 7.12.2 layout)
  const int row = ti + mr;           // this lane's A row
  const int col = tj + mr;           // this lane's B col

  // Preload all fragments -> loads pipeline, single dscnt wait before chain.
  float a0[8], a1[8], b0[8], b1[8];
#pragma unroll
  for (int k8 = 0; k8 < 8; ++k8) {
    const int k0 = k8 * 4 + kh;
    a0[k8] = A[row * LDA + k0];
    a1[k8] = A[row * LDA + k0 + 1];
    b0[k8] = B[k0 * LDA + col];
    b1[k8] = B[(k0 + 1) * LDA + col];
  }
  v8f acc = {0.f, 0.f, 0.f, 0.f, 0.f, 0.f, 0.f, 0.f};
#pragma unroll
  for (int k8 = 0; k8 < 8; ++k8) {
    const int k0 = k8 * 4 + kh;
    v2f a, b;
    if (AF_A) {
      a[0] = sa * a0[k8] + (row == k0     ? da : 0.f);
      a[1] = sa * a1[k8] + (row == k0 + 1 ? da : 0.f);
    } else {
      a[0] = a0[k8];
      a[1] = a1[k8];
    }
    if (AF_B) {
      b[0] = sb * b0[k8] + (k0 == col     ? db : 0.f);
      b[1] = sb * b1[k8] + (k0 + 1 == col ? db : 0.f);
    } else {
      b[0] = b0[k8];
      b[1] = b1[k8];
    }
    acc = __builtin_amdgcn_wmma_f32_16x16x4_f32(
        /*neg_a=*/false, a, /*neg_b=*/false, b,
        /*c_mod=*/(short)0, acc, /*reuse_a=*/false, /*reuse_b=*/false);
  }
  __syncthreads();  // every wave finished reading A/B before anyone writes D
  const int rbase = (lane >> 4) << 3;  // C/D rows: vgpr r + 8*(lane>=16)
#pragma unroll
  for (int r = 0; r < 8; ++r)
    D[(ti + rbase + r) * LDA + (tj + mr)] = acc[r];
  __syncthreads();
}

// Plain GEMM: D = A @ B (no VALU on the fragment path).
__device__ __forceinline__ void mm(const float* A, const float* B, float* D) {
  mmf<0, 0>(A, 1.f, 0.f, B, 1.f, 0.f, D);
}

// D = sc * S + dg * I   (element-wise; trailing barrier)
__device__ __forceinline__ void ew_saxd(float* D, const float* S, float sc,
                                        float dg) {
  for (int e = threadIdx.x; e < 1024; e += 128) {
    const int r = e >> 5, c = e & 31;
    D[r * LDA + c] = sc * S[r * LDA + c] + ((r == c) ? dg : 0.f);
  }
  __syncthreads();
}

__device__ __forceinline__ void ew_setI(float* D) {
  for (int e = threadIdx.x; e < 1024; e += 128) {
    const int r = e >> 5, c = e & 31;
    D[r * LDA + c] = (r == c) ? 1.f : 0.f;
  }
  __syncthreads();
}

__device__ __forceinline__ float frob_norm(const float* A, float* red) {
  if (threadIdx.x == 0) red[0] = 0.f;
  __syncthreads();
  float p = 0.f;
  for (int e = threadIdx.x; e < 1024; e += 128) {
    const float v = A[(e >> 5) * LDA + (e & 31)];
    p += v * v;
  }
  atomicAdd(red, p);  // ds_add_f32
  __syncthreads();
  const float g = sqrtf(red[0]);
  __syncthreads();
  return g;
}

__device__ __forceinline__ void g2s(float* D, const float* __restrict__ G) {
  for (int e = threadIdx.x; e < 1024; e += 128)
    D[(e >> 5) * LDA + (e & 31)] = G[e];
  __syncthreads();
}

__device__ __forceinline__ void s2gs(float* __restrict__ G, const float* S,
                                     float sc) {
  for (int e = threadIdx.x; e < 1024; e += 128)
    G[e] = sc * S[(e >> 5) * LDA + (e & 31)];
  __syncthreads();
}

// Coupled Newton-Schulz: Ain (normalized SPD) -> sY = Ain^{1/2}, sZ = Ain^{-1/2}.
// Ain may alias sY (copy skipped).
__device__ __forceinline__ void ns_sqrt(const float* Ain, float* sY, float* sZ,
                                        float* sU, int iters) {
  if (Ain != sY) ew_saxd(sY, Ain, 1.f, 0.f);
  ew_setI(sZ);
  for (int it = 0; it < iters; ++it) {
    mm(sZ, sY, sU);                                // U = Z@Y
    mmf<0, 1>(sY, 1.f, 0.f, sU, -0.5f, 1.5f, sY);  // Y = Y@(1.5I - 0.5U)
    mmf<1, 0>(sU, -0.5f, 1.5f, sZ, 1.f, 0.f, sZ);  // Z = (1.5I - 0.5U)@Z
  }
}

// ---- kernel 1: per (c,n) centroid slice, compute C^{-1/2}, C^{+1/2} --------
__global__ void __launch_bounds__(128)
k_prep(const float* __restrict__ C, float* __restrict__ Cm05,
       float* __restrict__ Cp05) {
  __shared__ float sA[MSZ], sY[MSZ], sZ[MSZ], sU[MSZ];
  __shared__ float red[1];
  const int slice = blockIdx.x;  // 0..63 == c*16 + n
  g2s(sA, C + slice * 1024);
  const float g = frob_norm(sA, red);
  ew_saxd(sA, sA, 1.f / g, 0.f);
  ns_sqrt(sA, sY, sZ, sU, 14);
  const float sg = sqrtf(g);
  s2gs(Cp05 + slice * 1024, sY, sg);        // C^{+1/2}
  s2gs(Cm05 + slice * 1024, sZ, 1.f / sg);  // C^{-1/2}
}

// ---- kernel 2: per gathered sample slice, logm + scaled accumulation ------
__global__ void __launch_bounds__(128)
k_log_acc(const float* __restrict__ X, const int* __restrict__ idx,
          const float* __restrict__ Cm05, float* __restrict__ Lacc) {
  __shared__ float sA[MSZ], sCm[MSZ], sY[MSZ], sZ[MSZ], sU[MSZ];
  __shared__ float red[1];
  const int b = blockIdx.x >> 5;      // 0..1023
  const int slice = blockIdx.x & 31;  // c*16 + n
  const int src = idx[b];
  g2s(sA, X + ((size_t)src * 32 + slice) * 1024);
  g2s(sCm, Cm05 + slice * 1024);
  mm(sCm, sA, sU);   // U = C^{-1/2} X
  mm(sU, sCm, sA);   // A = M = C^{-1/2} X C^{-1/2}
  const float g = frob_norm(sA, red);
  ew_saxd(sA, sA, 1.f / g, 0.f);  // eigenvalues in (0,1]
  // inverse scaling-and-squaring: two inverse-free square roots
  ns_sqrt(sA, sY, sZ, sU, 6);  // sY = M'^{1/2}
  ns_sqrt(sY, sY, sZ, sU, 6);  // sY = R = M'^{1/4} (in place)
  // 4th-order Taylor log, W = R - I fused:  W(I - W(I/2 - W(I/3 - W/4)))
  mmf<1, 1>(sY, 1.f, -1.f, sY, -0.25f, 7.f / 12.f, sU);  // U = W@((1/3)I - W/4)
  mmf<1, 1>(sY, 1.f, -1.f, sU, -1.f, 0.5f, sZ);          // Z = W@(0.5I - U)
  mmf<1, 1>(sY, 1.f, -1.f, sZ, -1.f, 1.f, sU);           // U = W@(I-Z) = log R
  const float lg = logf(g);
  float* acc = Lacc + slice * 1024;
  for (int e = threadIdx.x; e < 1024; e += 128) {
    const int r = e >> 5, c = e & 31;
    const float v = 4.f * sU[r * LDA + c] + ((r == c) ? lg : 0.f);
    atomicAdd(&acc[e], v * ETA_OVER_B);  // global_atomic_add_f32
  }
}

// ---- kernel 3: expm(L_m) and final congruence -----------------------------
__global__ void __launch_bounds__(128)
k_finish(const float* __restrict__ Lacc, const float* __restrict__ Cp05,
         float* __restrict__ out) {
  __shared__ float sA[MSZ], sH[MSZ], sU[MSZ], sCp[MSZ];
  const int slice = blockIdx.x;  // 0..63
  g2s(sA, Lacc + slice * 1024);
  ew_saxd(sA, sA, 1.f / 16.f, 0.f);  // scaling (s = 4)
  // Horner Taylor, "I +" terms fused into the B operand of the next GEMM:
  mmf<0, 1>(sA, 1.f, 0.f, sA, 1.f / 6.f, 1.f, sU);  // U = A@(A/6 + I)
#pragma unroll
  for (int k = 5; k >= 2; --k)
    mmf<0, 1>(sA, 1.f, 0.f, sU, 1.f / (float)k, 1.f, sU);  // U = A@(U/k + I)
  // E = I + U; four squarings, the first fused as (I+U)@(I+U):
  mmf<1, 1>(sU, 1.f, 1.f, sU, 1.f, 1.f, sH);
#pragma unroll
  for (int s = 0; s < 3; ++s)
    mm(sH, sH, sH);
  g2s(sCp, Cp05 + slice * 1024);
  mm(sCp, sH, sU);  // C^{1/2} E
  mm(sU, sCp, sH);  // C^{1/2} E C^{1/2}
  s2gs(out + slice * 1024, sH, 1.f);
}

// ---------------------------------------------------------------------------
extern "C" void kernel_launch(void* const* d_in, const int* in_sizes, int n_in,
                              void* d_out, int out_size, void* d_ws,
                              size_t ws_size, hipStream_t stream) {
  (void)in_sizes; (void)n_in; (void)out_size; (void)ws_size;
  const float* X = (const float*)d_in[0];  // [2048,2,16,32,32]
  const float* C = (const float*)d_in[1];  // [2,16,32,32]
  const int* idx = (const int*)d_in[2];    // [1024]
  float* out = (float*)d_out;              // [2,16,32,32]
  float* ws = (float*)d_ws;
  float* Cm05 = ws;                  // 64 * 1024 floats
  float* Cp05 = ws + 64 * 1024;      // 64 * 1024 floats
  float* Lacc = ws + 2 * 64 * 1024;  // 64 * 1024 floats (accumulator)

  hipMemsetAsync(Lacc, 0, 64 * 1024 * sizeof(float), stream);
  k_prep<<<64, 128, 0, stream>>>(C, Cm05, Cp05);
  k_log_acc<<<32768, 128, 0, stream>>>(X, idx, Cm05, Lacc);
  k_finish<<<64, 128, 0, stream>>>(Lacc, Cp05, out);
}